// GAT_39298950758976
// MI455X (gfx1250) — compile-verified
//
#include <hip/hip_runtime.h>
#include <hip/hip_bf16.h>
#include <stdint.h>

// ---------------------------------------------------------------------------
// GAT forward for MI455X (gfx1250, wave32).
//   h = x @ W1                        -> WMMA f32 16x16x4 (exact fp32 path)
//     W1 staged into LDS via Tensor Data Mover (tensor_load_to_lds, TENSORcnt)
//   as = h.att_src ; ad = h.att_dst   -> fused into GEMM via LDS atomics
//   edge softmax (segment max/sum)    -> hw uint atomicMax (order-encoded) + f32 atomicAdd
//   out[dst] += w * h[src] ; out+=b1  -> 64-thread-per-edge coalesced f32 atomics
//   y[g] = sigmoid(relu(out).flatten(g) . Wlin + blin)
// ---------------------------------------------------------------------------

#define IN_C   151
#define K_PAD  152
#define HID    64
#define NPG    128

typedef float    v2f   __attribute__((ext_vector_type(2)));
typedef float    v8f   __attribute__((ext_vector_type(8)));
typedef unsigned u32x4 __attribute__((ext_vector_type(4)));
typedef unsigned u32x8 __attribute__((ext_vector_type(8)));

__device__ __forceinline__ unsigned f2ord(float f) {
    unsigned b = __float_as_uint(f);
    return (b & 0x80000000u) ? ~b : (b | 0x80000000u);
}
__device__ __forceinline__ float ord2f(unsigned k) {
    unsigned b = (k & 0x80000000u) ? (k & 0x7fffffffu) : ~k;
    return __uint_as_float(b);
}

// ---------------------------------------------------------------------------
// K1: h = x @ W1 via V_WMMA_F32_16X16X4_F32.  2 waves / block, 32 rows / block.
// W1 tile DMA'd into LDS by the Tensor Data Mover (issued once by wave 0).
// Also computes as[i] = h[i].att_src, ad[i] = h[i].att_dst.
// ---------------------------------------------------------------------------
__global__ __launch_bounds__(64) void gat_gemm_wmma(
    const float* __restrict__ x, const float* __restrict__ W1,
    const float* __restrict__ attS, const float* __restrict__ attD,
    float* __restrict__ h, float* __restrict__ as_, float* __restrict__ ad_)
{
    __shared__ float Wlds[K_PAD * HID];        // 9728 f  (38 KB)
    __shared__ float Xlds[2 * 16 * K_PAD];     // 4864 f  (19 KB)
    __shared__ float aS[HID], aD[HID];
    __shared__ float asAcc[32], adAcc[32];

    const int tid  = threadIdx.x;       // 0..63
    const int lane = tid & 31;
    const int wave = tid >> 5;          // 0..1
    const int rowWave = blockIdx.x * 32 + wave * 16;

    // --- TDM: DMA W1 (151*64 contiguous f32) into LDS, wave 0 only ---------
    if (wave == 0) {
        const unsigned n_elem  = (unsigned)(IN_C * HID);           // 9664
        const unsigned ldsOff  = (unsigned)(uintptr_t)(void*)&Wlds[0];
        const uint64_t gaddr   = (uint64_t)(uintptr_t)(const void*)W1;

        u32x4 g0;
        g0.x = 0x1u;                                   // count=1, user mode
        g0.y = ldsOff;                                 // lds_addr [63:32]
        g0.z = (unsigned)(gaddr & 0xFFFFFFFFu);        // global_addr lo
        g0.w = (unsigned)((gaddr >> 32) & 0x01FFFFFFu) // global_addr [56:32]
               | 0x80000000u;                          // type=2 ("image")

        u32x8 g1;
        g1.s0 = (2u << 16);                            // data_size = 4B
        g1.s1 = (n_elem & 0xFFFFu) << 16;              // tensor_dim0 [15:0]
        g1.s2 = (n_elem >> 16) | (1u << 16);           // tensor_dim0 hi | tensor_dim1=1
        g1.s3 = (n_elem & 0xFFFFu) << 16;              // tile_dim0 = 9664
        g1.s4 = 1u;                                    // tile_dim1 = 1 (tile_dim2=0)
        g1.s5 = n_elem;                                // tensor_dim0_stride lo
        g1.s6 = 0u;                                    // stride hi / dim1_stride lo
        g1.s7 = 0u;

        asm volatile("tensor_load_to_lds %0, %1"
                     :: "s"(g0), "s"(g1) : "memory");
        __builtin_amdgcn_s_wait_tensorcnt(0);
    }

    // --- K-pad row of W + attention vectors (outside the TDM target) -------
    for (int i = tid; i < HID; i += 64) {
        Wlds[IN_C * HID + i] = 0.0f;
        aS[i] = attS[i];
        aD[i] = attD[i];
    }
    if (tid < 32) { asAcc[tid] = 0.0f; adAcc[tid] = 0.0f; }

    // --- stage 16x151 x-tile per wave (contiguous span -> coalesced) -------
    const float* xw = x + (size_t)rowWave * IN_C;
    float* Xw = Xlds + wave * 16 * K_PAD;
    for (int i = lane; i < 16 * IN_C; i += 32) {
        int r = i / IN_C;
        int c = i - r * IN_C;
        Xw[r * K_PAD + c] = xw[i];
    }
    for (int r = lane; r < 16; r += 32) Xw[r * K_PAD + IN_C] = 0.0f;
    __syncthreads();

    // --- WMMA accumulation: 38 K-chunks of 4, 4 N-tiles of 16 --------------
    const int lr = lane & 15;
    const int hi = lane >> 4;                  // lane half selects K pair
    v8f acc[4] = {};
#pragma unroll 1
    for (int kc = 0; kc < K_PAD / 4; ++kc) {
        const int kk = kc * 4 + hi * 2;
        v2f a;
        a.x = Xw[lr * K_PAD + kk];
        a.y = Xw[lr * K_PAD + kk + 1];
        const float* w0 = Wlds + kk * HID + lr;
        const float* w1 = w0 + HID;
#pragma unroll
        for (int t = 0; t < 4; ++t) {
            v2f b;
            b.x = w0[t * 16];
            b.y = w1[t * 16];
            acc[t] = __builtin_amdgcn_wmma_f32_16x16x4_f32(
                false, a, false, b, (short)0, acc[t], false, false);
        }
    }

    // --- write h + reduce attention dots (C layout: vgpr r -> row r / r+8) -
#pragma unroll
    for (int t = 0; t < 4; ++t) {
#pragma unroll
        for (int r = 0; r < 8; ++r) {
            const int row = hi ? (r + 8) : r;
            const int col = t * 16 + lr;
            const float v = acc[t][r];
            h[(size_t)(rowWave + row) * HID + col] = v;
            atomicAdd(&asAcc[wave * 16 + row], v * aS[col]);
            atomicAdd(&adAcc[wave * 16 + row], v * aD[col]);
        }
    }
    __syncthreads();
    if (tid < 32) {
        as_[blockIdx.x * 32 + tid] = asAcc[tid];
        ad_[blockIdx.x * 32 + tid] = adAcc[tid];
    }
}

// ---------------------------------------------------------------------------
// K0: out = b1 broadcast, amax keys = 0 (== -inf encoded), denom = 0
// ---------------------------------------------------------------------------
__global__ void gat_init(float* __restrict__ out, unsigned* __restrict__ amaxk,
                         float* __restrict__ denom, const float* __restrict__ b1,
                         int n_nodes)
{
    size_t i = (size_t)blockIdx.x * blockDim.x + threadIdx.x;
    size_t total = (size_t)n_nodes * HID;
    if (i < total) out[i] = b1[i & (HID - 1)];
    if (i < (size_t)n_nodes) { amaxk[i] = 0u; denom[i] = 0.0f; }
}

// ---------------------------------------------------------------------------
// K2: segment max of leaky-relu edge scores (hardware u32 atomic max)
// ---------------------------------------------------------------------------
__global__ void gat_edge_max(const int* __restrict__ src, const int* __restrict__ dst,
                             const float* __restrict__ as_, const float* __restrict__ ad_,
                             unsigned* __restrict__ amaxk, int E, int Etot)
{
    int e = blockIdx.x * blockDim.x + threadIdx.x;
    if (e >= Etot) return;
    int s, d;
    if (e < E) { s = src[e]; d = dst[e]; } else { s = d = e - E; }
    float a = as_[s] + ad_[d];
    a = a > 0.0f ? a : 0.2f * a;
    atomicMax(amaxk + d, f2ord(a));
}

// ---------------------------------------------------------------------------
// K3: ex = exp(alpha - amax[dst]); denom[dst] += ex
// ---------------------------------------------------------------------------
__global__ void gat_edge_exp(const int* __restrict__ src, const int* __restrict__ dst,
                             const float* __restrict__ as_, const float* __restrict__ ad_,
                             const unsigned* __restrict__ amaxk, float* __restrict__ exv,
                             float* __restrict__ denom, int E, int Etot)
{
    int e = blockIdx.x * blockDim.x + threadIdx.x;
    if (e >= Etot) return;
    int s, d;
    if (e < E) { s = src[e]; d = dst[e]; } else { s = d = e - E; }
    float a = as_[s] + ad_[d];
    a = a > 0.0f ? a : 0.2f * a;
    float ex = __expf(a - ord2f(amaxk[d]));
    exv[e] = ex;
    atomicAdd(&denom[d], ex);
}

// ---------------------------------------------------------------------------
// K4: out[dst] += (ex/denom[dst]) * h[src]  (64 threads = 1 edge, 1/channel)
// ---------------------------------------------------------------------------
__global__ void gat_edge_aggr(const int* __restrict__ src, const int* __restrict__ dst,
                              const float* __restrict__ exv, const float* __restrict__ denom,
                              const float* __restrict__ h, float* __restrict__ out,
                              int E, int Etot)
{
    long long gidx = (long long)blockIdx.x * blockDim.x + threadIdx.x;
    int e = (int)(gidx >> 6);
    int c = (int)(gidx & 63);
    if (e >= Etot) return;
    int s, d;
    if (e < E) { s = src[e]; d = dst[e]; } else { s = d = e - E; }
    float w = exv[e] / denom[d];
    atomicAdd(&out[(size_t)d * HID + c], w * h[(size_t)s * HID + c]);
}

// ---------------------------------------------------------------------------
// K5: per-graph readout:  sigmoid( relu(out[g]).flatten() . Wlin + blin )
// ---------------------------------------------------------------------------
__global__ __launch_bounds__(256) void gat_readout(
    const float* __restrict__ out, const float* __restrict__ Wlin,
    const float* __restrict__ blin, float* __restrict__ y)
{
    __shared__ float red[256];
    const int g = blockIdx.x;
    const int tid = threadIdx.x;
    const float* row = out + (size_t)g * (NPG * HID);
    float acc = 0.0f;
#pragma unroll 4
    for (int j = tid; j < NPG * HID; j += 256) {
        float v = row[j];
        v = v > 0.0f ? v : 0.0f;
        acc += v * Wlin[j];
    }
    red[tid] = acc;
    __syncthreads();
    for (int s = 128; s > 0; s >>= 1) {
        if (tid < s) red[tid] += red[tid + s];
        __syncthreads();
    }
    if (tid == 0) {
        float logit = red[0] + blin[0];
        y[g] = 1.0f / (1.0f + __expf(-logit));
    }
}

// ---------------------------------------------------------------------------
extern "C" void kernel_launch(void* const* d_in, const int* in_sizes, int n_in,
                              void* d_out, int out_size, void* d_ws, size_t ws_size,
                              hipStream_t stream)
{
    const float* x    = (const float*)d_in[0];
    const int*   ei   = (const int*)  d_in[1];
    const float* W1   = (const float*)d_in[2];
    const float* attS = (const float*)d_in[3];
    const float* attD = (const float*)d_in[4];
    const float* b1   = (const float*)d_in[5];
    const float* Wlin = (const float*)d_in[6];
    const float* blin = (const float*)d_in[7];
    float* y = (float*)d_out;

    const int N    = in_sizes[0] / IN_C;       // 262144 nodes
    const int E    = in_sizes[1] / 2;          // 2097152 edges
    const int Etot = E + N;                    // + self loops
    const int G    = out_size;                 // 2048 graphs

    const int* src = ei;
    const int* dst = ei + E;

    // ---- workspace carve-up (bytes) ----
    char* ws = (char*)d_ws;
    size_t off = 0;
    auto carve = [&](size_t bytes) { char* p = ws + off; off += (bytes + 255) & ~(size_t)255; return p; };
    float*    h     = (float*)   carve((size_t)N * HID * sizeof(float));
    float*    aggr  = (float*)   carve((size_t)N * HID * sizeof(float));
    float*    as_   = (float*)   carve((size_t)N * sizeof(float));
    float*    ad_   = (float*)   carve((size_t)N * sizeof(float));
    unsigned* amaxk = (unsigned*)carve((size_t)N * sizeof(unsigned));
    float*    denom = (float*)   carve((size_t)N * sizeof(float));
    float*    exv   = (float*)   carve((size_t)Etot * sizeof(float));
    (void)ws_size; (void)n_in;

    // K0: init out=b1, amax=-inf, denom=0
    {
        size_t total = (size_t)N * HID;
        int blocks = (int)((total + 255) / 256);
        gat_init<<<blocks, 256, 0, stream>>>(aggr, amaxk, denom, b1, N);
    }
    // K1: WMMA GEMM (TDM-staged W1) + attention dots
    gat_gemm_wmma<<<N / 32, 64, 0, stream>>>(x, W1, attS, attD, h, as_, ad_);
    // K2: segment max
    gat_edge_max<<<(Etot + 255) / 256, 256, 0, stream>>>(src, dst, as_, ad_, amaxk, E, Etot);
    // K3: exp + segment sum
    gat_edge_exp<<<(Etot + 255) / 256, 256, 0, stream>>>(src, dst, as_, ad_, amaxk, exv, denom, E, Etot);
    // K4: weighted aggregation
    {
        long long threads = (long long)Etot * HID;
        int blocks = (int)((threads + 255) / 256);
        gat_edge_aggr<<<blocks, 256, 0, stream>>>(src, dst, exv, denom, h, aggr, E, Etot);
    }
    // K5: per-graph readout
    gat_readout<<<G, 256, 0, stream>>>(aggr, Wlin, blin, y);
}